// SelfBatchLoss_86096914415964
// MI455X (gfx1250) — compile-verified
//
#include <hip/hip_runtime.h>
#include <math.h>

// ---------------- problem constants (match reference) ----------------
#define ALPHA_C 40.0f
#define BETA_C  2.0f
#define GAMMA_C 4.0f          // 0.1 * ALPHA
#define N_C     4096
#define D_C     512
#define WG_ROWS 128           // rows per workgroup (8 waves x 16)
#define TS      16            // WMMA tile edge
#define STRIDE  516           // padded floats per staged row (bank-conflict free)

typedef __attribute__((ext_vector_type(2))) float v2f;
typedef __attribute__((ext_vector_type(8))) float v8f;

// ---- CDNA5 async global->LDS staging (ASYNCcnt-tracked, CDNA5 ISA) ----
__device__ __forceinline__ unsigned lds_offset(const void* p) {
    // generic pointer to LDS: low 32 bits are the LDS byte offset
    return (unsigned)(unsigned long long)p;
}

__device__ __forceinline__ void async_load_lds_b128(unsigned lds_off, const void* gaddr) {
    asm volatile("global_load_async_to_lds_b128 %0, %1, off"
                 :: "v"(lds_off), "v"(gaddr)
                 : "memory");
}

__device__ __forceinline__ void wait_asynccnt0() {
    asm volatile("s_wait_asynccnt 0x0" ::: "memory");
}

// Streaming logsumexp of exp(-ALPHA*(x - beta)) tracking m = min(sim) (hardest positive).
__device__ __forceinline__ void lse_min_update(float& m, float& s, float x) {
    if (x >= m) {
        s += __expf(-ALPHA_C * (x - m));
    } else {
        s = s * __expf(-ALPHA_C * (m - x)) + 1.0f;
        m = x;
    }
}

// Streaming logsumexp of exp(GAMMA*x) tracking m = max(sim) (hardest negative).
__device__ __forceinline__ void lse_max_update(float& m, float& s, float x) {
    if (x <= m) {
        s += __expf(GAMMA_C * (x - m));
    } else {
        s = s * __expf(GAMMA_C * (m - x)) + 1.0f;
        m = x;
    }
}

__device__ __forceinline__ void combine_min(float& m, float& s, float m2, float s2) {
    float mm = fminf(m, m2);
    if (mm < INFINITY) {   // guard inf-inf -> NaN when both halves empty
        s = s * __expf(-ALPHA_C * (m - mm)) + s2 * __expf(-ALPHA_C * (m2 - mm));
        m = mm;
    }
}

__device__ __forceinline__ void combine_max(float& m, float& s, float m2, float s2) {
    float mm = fmaxf(m, m2);
    if (mm > -INFINITY) {
        s = s * __expf(GAMMA_C * (m - mm)) + s2 * __expf(GAMMA_C * (m2 - mm));
        m = mm;
    }
}

// ---------------------------------------------------------------------
// Kernel 1: fused X X^T + masked per-row streaming reductions.
// Each workgroup: 128 rows staged in LDS once (async DMA); loop over all
// 256 column tiles (async-staged); each wave computes one 16x16 sim tile
// via 128 fp32 WMMAs and folds it into streaming logsumexp state.
// ---------------------------------------------------------------------
__global__ __launch_bounds__(256) void simloss_main(const float* __restrict__ X,
                                                    const int* __restrict__ T,
                                                    float* __restrict__ rowloss,
                                                    float* __restrict__ rowdbg) {
    extern __shared__ float smem[];
    float* As = smem;                       // WG_ROWS x STRIDE
    float* Bs = smem + WG_ROWS * STRIDE;    // TS x STRIDE

    const int tid  = threadIdx.x;
    const int wave = tid >> 5;
    const int lane = tid & 31;
    const int h    = lane >> 4;             // K sub-offset selector (ISA lane halves)
    const int n    = lane & 15;             // A row / B column within tile
    const int rowbase = blockIdx.x * WG_ROWS;

    // Stage the 128x512 A-panel via async DMA (reused for all 256 column tiles).
#pragma unroll
    for (int i = 0; i < (WG_ROWS * D_C / 4) / 256; ++i) {   // 64 b128 per thread
        int q  = i * 256 + tid;
        int r  = q >> 7;                                    // 128 float4 per row
        int c4 = q & 127;
        async_load_lds_b128(lds_offset(As + r * STRIDE + c4 * 4),
                            X + (size_t)(rowbase + r) * D_C + c4 * 4);
    }

    // Class labels for the 8 rows this lane accumulates (rows 8h + r of the wave tile).
    int trow[8];
#pragma unroll
    for (int r = 0; r < 8; ++r) trow[r] = T[rowbase + wave * TS + 8 * h + r];

    float mp[8], sp[8], mn[8], sn[8];
#pragma unroll
    for (int r = 0; r < 8; ++r) { mp[r] = INFINITY; sp[r] = 0.0f; mn[r] = -INFINITY; sn[r] = 0.0f; }

    const float* arow = As + (wave * TS + n) * STRIDE + 2 * h;

    for (int ct = 0; ct < N_C / TS; ++ct) {
        const int colbase = ct * TS;

        __syncthreads();   // previous tile fully consumed by all waves

        // Stage the 16x512 B-panel straight into LDS (ASYNCcnt path).
#pragma unroll
        for (int i = 0; i < 8; ++i) {                       // 2048 float4 / 256 threads
            int q  = i * 256 + tid;
            int r  = q >> 7;
            int c4 = q & 127;
            async_load_lds_b128(lds_offset(Bs + r * STRIDE + c4 * 4),
                                X + (size_t)(colbase + r) * D_C + c4 * 4);
        }
        wait_asynccnt0();    // this wave's async transfers landed in LDS
        __syncthreads();     // all waves' transfers visible

        // Prefetch next tile's 32KB into cache while we compute (global_prefetch_b8).
        if (ct + 1 < N_C / TS) {
            const char* p = (const char*)(X + (size_t)(colbase + TS) * D_C);
            __builtin_prefetch(p + tid * 128, 0, 1);
        }

        const int    tcol = T[colbase + n];
        const float* brow = Bs + n * STRIDE + 2 * h;

        // 16x16 fp32 sim tile: D = A(16x512) * B^T via 128 x wmma_f32_16x16x4_f32.
        v8f acc = {};
#pragma unroll 16
        for (int k = 0; k < D_C; k += 4) {
            v2f a = *(const v2f*)(arow + k);
            v2f b = *(const v2f*)(brow + k);
            acc = __builtin_amdgcn_wmma_f32_16x16x4_f32(
                false, a, false, b, (short)0, acc, false, false);
        }

        // Epilogue: lane holds sim[row = 8h+r][col = n] in acc[r].
#pragma unroll
        for (int r = 0; r < 8; ++r) {
            float s = acc[r];
            if (trow[r] == tcol) {
                if (s < 1.0f) lse_min_update(mp[r], sp[r], s);   // positive (self excluded)
            } else {
                lse_max_update(mn[r], sn[r], s);                 // negative
            }
        }
    }

    // Cross-lane reduction over the 16 lanes of each half (rows differ per half).
#pragma unroll
    for (int r = 0; r < 8; ++r) {
        float m1 = mp[r], s1 = sp[r], m2 = mn[r], s2 = sn[r];
#pragma unroll
        for (int off = 1; off < 16; off <<= 1) {
            float om  = __shfl_xor(m1, off, 32);
            float os  = __shfl_xor(s1, off, 32);
            combine_min(m1, s1, om, os);
            float on  = __shfl_xor(m2, off, 32);
            float osn = __shfl_xor(s2, off, 32);
            combine_max(m2, s2, on, osn);
        }
        if (n == 0) {
            const int grow = rowbase + wave * TS + 8 * h + r;
            const bool has_pos = (m1 < INFINITY) && (s1 > 0.0f);
            const bool has_neg = (m2 > -INFINITY) && (s2 > 0.0f);
            const bool valid = has_pos && has_neg && (m1 - BETA_C <= m2);
            const float pl = __logf(s1) - ALPHA_C * (m1 - BETA_C);
            const float nl = __logf(s2) + GAMMA_C * m2;
            rowloss[grow] = valid ? (pl + nl) : 0.0f;
            rowdbg[grow]  = valid ? (m2 - m1 + BETA_C) : 0.0f;
        }
    }
}

// ---------------------------------------------------------------------
// Kernel 2: reduce per-row arrays + last-row pair statistics.
// ---------------------------------------------------------------------
__global__ __launch_bounds__(256) void simloss_final(const float* __restrict__ X,
                                                     const int* __restrict__ T,
                                                     const float* __restrict__ rowloss,
                                                     const float* __restrict__ rowdbg,
                                                     float* __restrict__ out) {
    __shared__ float xlast[D_C];
    __shared__ float red[256];
    const int tid = threadIdx.x;

    for (int i = tid; i < D_C; i += 256) xlast[i] = X[(size_t)(N_C - 1) * D_C + i];
    __syncthreads();

    float lsum = 0.0f, dsum = 0.0f;
    for (int i = tid; i < N_C; i += 256) { lsum += rowloss[i]; dsum += rowdbg[i]; }

    const int tlast = T[N_C - 1];
    float psum = 0.0f, nsum = 0.0f;
    float pcnt = 0.0f, ncnt = 0.0f;
    for (int j = tid; j < N_C; j += 256) {
        const float4* xj = (const float4*)(X + (size_t)j * D_C);
        float dot = 0.0f;
#pragma unroll 8
        for (int q = 0; q < D_C / 4; ++q) {
            float4 v = xj[q];
            const float4 w = *(const float4*)(xlast + q * 4);
            dot = fmaf(v.x, w.x, fmaf(v.y, w.y, fmaf(v.z, w.z, fmaf(v.w, w.w, dot))));
        }
        if (T[j] == tlast) {
            if (dot < 1.0f) { psum += dot; pcnt += 1.0f; }
        } else {
            nsum += dot; ncnt += 1.0f;
        }
    }

    auto block_reduce = [&](float v) -> float {
        red[tid] = v;
        __syncthreads();
        for (int s = 128; s > 0; s >>= 1) {
            if (tid < s) red[tid] += red[tid + s];
            __syncthreads();
        }
        float r = red[0];
        __syncthreads();
        return r;
    };

    const float L  = block_reduce(lsum);
    const float Dg = block_reduce(dsum);
    const float P  = block_reduce(psum);
    const float Pc = block_reduce(pcnt);
    const float Ns = block_reduce(nsum);
    const float Nc = block_reduce(ncnt);

    if (tid == 0) {
        out[0] = L / (float)N_C;                 // loss
        out[1] = Dg / (float)N_C;                // prec
        out[2] = P / fmaxf(Pc, 1.0f);            // mean_pos_sim (last row)
        out[3] = Ns / fmaxf(Nc, 1.0f);           // mean_neg_sim (last row)
    }
}

extern "C" void kernel_launch(void* const* d_in, const int* in_sizes, int n_in,
                              void* d_out, int out_size, void* d_ws, size_t ws_size,
                              hipStream_t stream) {
    const float* X = (const float*)d_in[0];
    const int*   T = (const int*)d_in[1];
    float* out = (float*)d_out;

    float* rowloss = (float*)d_ws;           // 4096 floats
    float* rowdbg  = rowloss + N_C;          // 4096 floats

    const size_t shmem = (size_t)(WG_ROWS * STRIDE + TS * STRIDE) * sizeof(float); // 297,216 B
    simloss_main<<<N_C / WG_ROWS, 256, shmem, stream>>>(X, T, rowloss, rowdbg);
    simloss_final<<<1, 256, 0, stream>>>(X, T, rowloss, rowdbg, out);
}